// exp_model_24902220382873
// MI455X (gfx1250) — compile-verified
//
#include <hip/hip_runtime.h>

// out_i = a*out_{i-1} + c  ->  out_i = K*a^i + L  (L = c/(1-a), K = out0 - L)
// Per 256-elem tile, a^(t0+16m+n) = a^t0 * (a^16)^m * a^n is a rank-1 outer
// product -> one V_WMMA_F32_16X16X4_F32 per tile produces 256 f32 outputs.
// Kernel is HBM-store-bound (80 MB @ 23.3 TB/s ~= 3.4 us floor), so the
// inner loop is kept to: f64 mul, f64->f32 cvt, WMMA, 8 unpredicated stores.

typedef float v2f __attribute__((ext_vector_type(2)));
typedef float v8f __attribute__((ext_vector_type(8)));

__device__ __forceinline__ float rfl_f32(float x) {
    // force scalar (SGPR) value -> uniform branches keep EXEC all-1s for WMMA
    return __uint_as_float((unsigned)__builtin_amdgcn_readfirstlane((int)__float_as_uint(x)));
}

__device__ __forceinline__ double dipow(double b, unsigned e) {
    double r = 1.0;
    while (e) {
        if (e & 1u) r *= b;
        b *= b;
        e >>= 1;
    }
    return r;
}

__global__ void exp_model_recurrence_kernel(const float* __restrict__ X,
                                            const float* __restrict__ w1,
                                            const float* __restrict__ b1,
                                            const float* __restrict__ b2,
                                            float* __restrict__ out, int N) {
    const float a   = rfl_f32(w1[0]);
    const float b1v = rfl_f32(b1[0]);
    const float b2v = rfl_f32(b2[0]);
    const float xv  = rfl_f32(X[0]);
    const float out0 = a * xv + b1v;
    const float c    = b1v + b2v;

    const int lane          = (int)(threadIdx.x & 31u);
    const int wavesPerBlock = (int)(blockDim.x >> 5);
    const int gwave         = (int)blockIdx.x * wavesPerBlock + (int)(threadIdx.x >> 5);
    const int nwaves        = (int)gridDim.x * wavesPerBlock;

    if (a == 1.0f) {
        // degenerate fixed point: out_i = out0 + c*i (uniform scalar branch)
        const int tid    = (int)(blockIdx.x * blockDim.x + threadIdx.x);
        const int stride = (int)(gridDim.x * blockDim.x);
        const double o0 = (double)out0, cd = (double)c;
        for (int i = tid; i < N; i += stride)
            out[i] = (float)(o0 + cd * (double)i);
        return;
    }

    const double ad = (double)a;
    const double fd = (double)c / (1.0 - ad);   // fixed point L
    const double Kd = (double)out0 - fd;        // K
    const float  Lf = (float)fd;

    const double a16d  = dipow(ad, 16u);
    const double a256d = dipow(ad, 256u);

    // B (4x16, K=0 row only): lanes 0..15 hold a^n, n = lane
    const float bl = (lane < 16) ? (float)dipow(ad, (unsigned)lane) : 0.0f;
    const v2f Bv = { bl, 0.0f };

    // A (16x4, K=0 col only): lanes 0..15 hold K * a^(16m), m = lane
    const double Abase = (lane < 16) ? Kd * dipow(a16d, (unsigned)lane) : 0.0;

    const int fullTiles = N >> 8;                  // tiles with all 256 in range
    double Pd   = dipow(a256d, (unsigned)gwave);   // a^(256*t) for first tile
    double step = dipow(a256d, (unsigned)nwaves);  // advance per grid-stride

    v8f Cv;
#pragma unroll
    for (int r = 0; r < 8; ++r) Cv[r] = Lf;

    const int laneoff = (lane & 15) + ((lane >> 4) << 7);

    for (int t = gwave; t < fullTiles; t += nwaves) {
        const v2f Av = { (float)(Abase * Pd), 0.0f };

        // D = A x B + C : 256 outputs of the recurrence in one instruction
        const v8f Dv = __builtin_amdgcn_wmma_f32_16x16x4_f32(
            /*neg_a=*/false, Av, /*neg_b=*/false, Bv,
            /*c_mod=*/(short)0, Cv, /*reuse_a=*/false, /*reuse_b=*/false);

        // C/D layout: VGPR r, lane l -> element (m = r + (l>=16 ? 8:0), n = l&15)
        // -> 8 unpredicated b32 stores, one shared addr VGPR + imm offsets
        const int base = (t << 8) + laneoff;
#pragma unroll
        for (int r = 0; r < 8; ++r)
            out[base + (r << 4)] = Dv[r];

        Pd *= step;
    }

    // Tail (< 256 elements; empty when N % 256 == 0): direct closed form.
    const int tailStart = fullTiles << 8;
    const int i = tailStart + (int)(blockIdx.x * blockDim.x + threadIdx.x);
    if (i < N)
        out[i] = (float)(Kd * dipow(ad, (unsigned)i) + fd);
}

extern "C" void kernel_launch(void* const* d_in, const int* in_sizes, int n_in,
                              void* d_out, int out_size, void* d_ws, size_t ws_size,
                              hipStream_t stream) {
    (void)in_sizes; (void)n_in; (void)d_ws; (void)ws_size;
    const float* X  = (const float*)d_in[0];
    const float* w1 = (const float*)d_in[1];
    const float* b1 = (const float*)d_in[2];
    const float* b2 = (const float*)d_in[3];
    float* out = (float*)d_out;
    const int N = out_size;

    const int wavesPerBlock = 8;           // 256 threads = 8 wave32s
    const int tiles = (N + 255) / 256;
    int blocks = (tiles + wavesPerBlock - 1) / wavesPerBlock;
    if (blocks > 2048) blocks = 2048;      // ~16K waves: enough to saturate HBM
    if (blocks < 1) blocks = 1;

    exp_model_recurrence_kernel<<<blocks, 256, 0, stream>>>(X, w1, b1, b2, out, N);
}